// UOT_loss_37065567765209
// MI455X (gfx1250) — compile-verified
//
#include <hip/hip_runtime.h>
#include <hip/hip_bf16.h>

#define NN 4096
#define DD 64
#define TR 16          // rows per block in streaming kernels
#define REG_M 5.0f
#define EPS_DEN 1e-16f

typedef float v2f __attribute__((ext_vector_type(2)));
typedef float v8f __attribute__((ext_vector_type(8)));

// ---------------------------------------------------------------------------
// Row norms ||x_i||^2, ||y_j||^2
// ---------------------------------------------------------------------------
__global__ void norms_k(const float* __restrict__ x, const float* __restrict__ y,
                        float* __restrict__ x2, float* __restrict__ y2) {
    int i = blockIdx.x * blockDim.x + threadIdx.x;
    if (i < NN) {
        const float* xr = x + (size_t)i * DD;
        const float* yr = y + (size_t)i * DD;
        float sx = 0.f, sy = 0.f;
        #pragma unroll 8
        for (int k = 0; k < DD; ++k) { sx += xr[k] * xr[k]; sy += yr[k] * yr[k]; }
        x2[i] = sx; y2[i] = sy;
    }
}

// ---------------------------------------------------------------------------
// Init: sumA/sumB reductions, r0_i = a_i*sumB, c0_j = b_j*sumA, zero c1,c2
// ---------------------------------------------------------------------------
__global__ void init_k(const float* __restrict__ a, const float* __restrict__ b,
                       float* __restrict__ r0, float* __restrict__ c0,
                       float* __restrict__ c1, float* __restrict__ c2) {
    __shared__ float sA[256], sB[256];
    int tid = threadIdx.x;
    float la = 0.f, lb = 0.f;
    for (int i = tid; i < NN; i += 256) { la += a[i]; lb += b[i]; }
    sA[tid] = la; sB[tid] = lb;
    __syncthreads();
    for (int s = 128; s > 0; s >>= 1) {
        if (tid < s) { sA[tid] += sA[tid + s]; sB[tid] += sB[tid + s]; }
        __syncthreads();
    }
    float sumA = sA[0], sumB = sB[0];
    for (int i = tid; i < NN; i += 256) {
        r0[i] = a[i] * sumB;   // rowsum of G0 = a_i * sum(b)
        c0[i] = b[i] * sumA;   // colsum of G0 = b_j * sum(a)
        c1[i] = 0.f;
        c2[i] = 0.f;
    }
}

// ---------------------------------------------------------------------------
// K = max(5a_i + 5b_j - M, 0) with M = max(x2_i + y2_j - 2 x.y, 0), and
// G0 = a_i*b_j.  Dot products via V_WMMA_F32_16X16X4_F32 (fp32 matrix core).
// One wave per 16x16 tile; 16 WMMA steps over k.
// ---------------------------------------------------------------------------
__global__ void kg_wmma_k(const float* __restrict__ x, const float* __restrict__ y,
                          const float* __restrict__ x2, const float* __restrict__ y2,
                          const float* __restrict__ a, const float* __restrict__ b,
                          float* __restrict__ Km, float* __restrict__ G) {
    int wave = (int)((blockIdx.x * blockDim.x + threadIdx.x) >> 5);
    int lane = threadIdx.x & 31;
    const int tiles_n = NN / 16;                 // 256
    int tm = wave / tiles_n, tn = wave % tiles_n;
    int row0 = tm * 16, col0 = tn * 16;
    int l = lane & 15, h = lane >> 4;

    // A 16x4 f32 layout: lanes 0-15 -> K={k0,k0+1}, lanes 16-31 -> K={k0+2,k0+3}
    const float* xp = x + (size_t)(row0 + l) * DD + 2 * h;
    // B 4x16 f32 layout (symmetric): per-lane float2 of y[col][k0+2h .. +1]
    const float* yp = y + (size_t)(col0 + l) * DD + 2 * h;

    v8f acc = {};
    #pragma unroll
    for (int k0 = 0; k0 < DD; k0 += 4) {
        v2f av = *(const v2f*)(xp + k0);
        v2f bv = *(const v2f*)(yp + k0);
        acc = __builtin_amdgcn_wmma_f32_16x16x4_f32(
            false, av, false, bv, (short)0, acc, false, false);
    }

    int j = col0 + l;
    float yj2 = y2[j];
    float bj  = b[j];
    #pragma unroll
    for (int v = 0; v < 8; ++v) {
        int i = row0 + v + 8 * h;               // C/D layout: VGPR v, lane-half h
        float m = x2[i] + yj2 - 2.0f * acc[v];
        m = fmaxf(m, 0.f);
        float kk = fmaxf(REG_M * a[i] + REG_M * bj - m, 0.f);
        size_t idx = (size_t)i * NN + j;
        Km[idx] = kk;
        G[idx]  = a[i] * bj;
    }
}

// ---------------------------------------------------------------------------
// One MM iteration: Gn = K*G / (5 r_i + 5 c_j + eps), in-place on G.
// Block owns TR=16 rows. Row sums -> LDS reduce -> r_next (no atomics).
// Col sums -> per-thread register accum over rows -> atomic f32 add c_next.
// Triple-buffered c: also zero our 16-entry chunk of c_zero (used 2 iters out).
// Working set G+K = 128MB, resident in 192MB L2.
// ---------------------------------------------------------------------------
__global__ void iter_k(float* __restrict__ G, const float* __restrict__ Km,
                       const float* __restrict__ r_cur, float* __restrict__ r_next,
                       const float* __restrict__ c_cur, float* __restrict__ c_next,
                       float* __restrict__ c_zero) {
    __shared__ float rs[TR];
    __shared__ float rowpart[TR * 256];
    __shared__ float r2[TR * 16];
    int tid = threadIdx.x;
    int row0 = blockIdx.x * TR;

    if (tid < TR) rs[tid] = REG_M * r_cur[row0 + tid];

    float4 cden[4];
    #pragma unroll
    for (int p = 0; p < 4; ++p) {
        float4 c = *(const float4*)(c_cur + p * 1024 + 4 * tid);
        cden[p].x = REG_M * c.x + EPS_DEN;
        cden[p].y = REG_M * c.y + EPS_DEN;
        cden[p].z = REG_M * c.z + EPS_DEN;
        cden[p].w = REG_M * c.w + EPS_DEN;
    }
    __syncthreads();

    float4 cacc[4];
    #pragma unroll
    for (int p = 0; p < 4; ++p) cacc[p] = make_float4(0.f, 0.f, 0.f, 0.f);

    for (int il = 0; il < TR; ++il) {
        size_t rowoff = (size_t)(row0 + il) * NN;
        float gr = rs[il];
        float lsum = 0.f;
        #pragma unroll
        for (int p = 0; p < 4; ++p) {
            size_t idx = rowoff + p * 1024 + 4 * tid;
            float4 g = *(const float4*)(G  + idx);
            float4 k = *(const float4*)(Km + idx);
            float4 gn;
            gn.x = k.x * g.x * __builtin_amdgcn_rcpf(gr + cden[p].x);
            gn.y = k.y * g.y * __builtin_amdgcn_rcpf(gr + cden[p].y);
            gn.z = k.z * g.z * __builtin_amdgcn_rcpf(gr + cden[p].z);
            gn.w = k.w * g.w * __builtin_amdgcn_rcpf(gr + cden[p].w);
            *(float4*)(G + idx) = gn;
            lsum += (gn.x + gn.y) + (gn.z + gn.w);
            cacc[p].x += gn.x; cacc[p].y += gn.y;
            cacc[p].z += gn.z; cacc[p].w += gn.w;
        }
        rowpart[il * 256 + tid] = lsum;
    }
    __syncthreads();

    // Row-sum reduction: 16 threads per row sum 16 entries each, then 16->1.
    {
        int row = tid >> 4, seg = tid & 15;
        float s = 0.f;
        #pragma unroll
        for (int q = 0; q < 16; ++q) s += rowpart[row * 256 + seg * 16 + q];
        r2[row * 16 + seg] = s;
    }
    __syncthreads();
    if (tid < TR) {
        float t = 0.f;
        #pragma unroll
        for (int q = 0; q < 16; ++q) t += r2[tid * 16 + q];
        r_next[row0 + tid] = t;
    }

    // Column partials: each column owned by exactly one thread in this block.
    #pragma unroll
    for (int p = 0; p < 4; ++p) {
        int j = p * 1024 + 4 * tid;
        atomicAdd(&c_next[j + 0], cacc[p].x);
        atomicAdd(&c_next[j + 1], cacc[p].y);
        atomicAdd(&c_next[j + 2], cacc[p].z);
        atomicAdd(&c_next[j + 3], cacc[p].w);
    }

    // Zero the buffer used two iterations from now (no one reads it this iter).
    if (tid < 16) c_zero[blockIdx.x * 16 + tid] = 0.f;
}

// ---------------------------------------------------------------------------
// Finalization
// ---------------------------------------------------------------------------
__global__ void zero_out_k(float* __restrict__ out) {
    if (threadIdx.x == 0) out[0] = 0.f;
}

// ot_loss = sum G*M, with M = 5a+5b-K where K>0 (G==0 wherever K==0).
__global__ void otloss_k(const float* __restrict__ G, const float* __restrict__ Km,
                         const float* __restrict__ a, const float* __restrict__ b,
                         float* __restrict__ out) {
    __shared__ float part[256];
    int tid = threadIdx.x;
    int row0 = blockIdx.x * TR;
    float4 bv[4];
    #pragma unroll
    for (int p = 0; p < 4; ++p) {
        float4 t = *(const float4*)(b + p * 1024 + 4 * tid);
        bv[p].x = REG_M * t.x; bv[p].y = REG_M * t.y;
        bv[p].z = REG_M * t.z; bv[p].w = REG_M * t.w;
    }
    float local = 0.f;
    for (int il = 0; il < TR; ++il) {
        size_t rowoff = (size_t)(row0 + il) * NN;
        float ai5 = REG_M * a[row0 + il];
        #pragma unroll
        for (int p = 0; p < 4; ++p) {
            size_t idx = rowoff + p * 1024 + 4 * tid;
            float4 g = *(const float4*)(G  + idx);
            float4 k = *(const float4*)(Km + idx);
            if (k.x > 0.f) local += g.x * (ai5 + bv[p].x - k.x);
            if (k.y > 0.f) local += g.y * (ai5 + bv[p].y - k.y);
            if (k.z > 0.f) local += g.z * (ai5 + bv[p].z - k.z);
            if (k.w > 0.f) local += g.w * (ai5 + bv[p].w - k.w);
        }
    }
    part[tid] = local;
    __syncthreads();
    for (int s = 128; s > 0; s >>= 1) {
        if (tid < s) part[tid] += part[tid + s];
        __syncthreads();
    }
    if (tid == 0) atomicAdd(&out[0], part[0]);
}

__global__ void marg_k(const float* __restrict__ r_final, const float* __restrict__ a,
                       float* __restrict__ out) {
    __shared__ float part[256];
    int tid = threadIdx.x;
    float local = 0.f;
    for (int i = tid; i < NN; i += 256) {
        float d = r_final[i] - a[i];
        local += d * d;
    }
    part[tid] = local;
    __syncthreads();
    for (int s = 128; s > 0; s >>= 1) {
        if (tid < s) part[tid] += part[tid + s];
        __syncthreads();
    }
    if (tid == 0) out[1] = part[0] * (1.0f / NN);
}

// ---------------------------------------------------------------------------
extern "C" void kernel_launch(void* const* d_in, const int* in_sizes, int n_in,
                              void* d_out, int out_size, void* d_ws, size_t ws_size,
                              hipStream_t stream) {
    const float* x = (const float*)d_in[0];   // source  [4096,64]
    const float* y = (const float*)d_in[1];   // target  [4096,64]
    const float* a = (const float*)d_in[2];   // source_density [4096]
    const float* b = (const float*)d_in[3];   // target_density [4096]
    float* out = (float*)d_out;               // [ot_loss, marginal_loss]

    char* ws = (char*)d_ws;
    const size_t mat_bytes = (size_t)NN * NN * sizeof(float);
    float* G  = (float*)(ws);
    float* Km = (float*)(ws + mat_bytes);
    float* x2 = (float*)(ws + 2 * mat_bytes);
    float* y2 = x2 + NN;
    float* r  = y2 + NN;   // 2 buffers of NN (ping-pong row sums)
    float* c  = r + 2 * NN; // 3 buffers of NN (triple-buffered col sums)

    norms_k<<<NN / 256, 256, 0, stream>>>(x, y, x2, y2);
    init_k<<<1, 256, 0, stream>>>(a, b, r, c, c + NN, c + 2 * NN);

    // 65536 tiles of 16x16, one wave per tile, 8 waves (256 threads) per block.
    kg_wmma_k<<<(NN / 16) * (NN / 16) / 8, 256, 0, stream>>>(x, y, x2, y2, a, b, Km, G);

    // fp32 stopThr=1e-15 is unreachable -> always the full 1000 iterations.
    for (int t = 0; t < 1000; ++t) {
        float* rc = r + (size_t)(t & 1) * NN;
        float* rn = r + (size_t)((t + 1) & 1) * NN;
        float* cc = c + (size_t)(t % 3) * NN;
        float* cn = c + (size_t)((t + 1) % 3) * NN;
        float* cz = c + (size_t)((t + 2) % 3) * NN;
        iter_k<<<NN / TR, 256, 0, stream>>>(G, Km, rc, rn, cc, cn, cz);
    }

    zero_out_k<<<1, 32, 0, stream>>>(out);
    otloss_k<<<NN / TR, 256, 0, stream>>>(G, Km, a, b, out);
    // After 1000 iterations final row sums live in r buffer ((999+1)&1) == 0.
    marg_k<<<1, 256, 0, stream>>>(r, a, out);
}